// NeuralMemory_81389630259300
// MI455X (gfx1250) — compile-verified
//
#include <hip/hip_runtime.h>

// Problem constants (fixed by the reference).
#define N_SLOTS 512
#define BATCH   1024
#define READ    128

typedef float v4f __attribute__((ext_vector_type(4)));

// ---- CDNA5 async global->LDS path (gfx1250), with safe fallback ------------
#if defined(__AMDGCN__) && __has_builtin(__builtin_amdgcn_global_load_async_to_lds_b32)
#define HAVE_ASYNC_LDS 1
#else
#define HAVE_ASYNC_LDS 0
#endif

typedef __attribute__((address_space(1))) int g_i32;
typedef __attribute__((address_space(3))) int l_i32;

__device__ __forceinline__ void async_copy_f32(const float* g, float* l) {
#if HAVE_ASYNC_LDS
  __builtin_amdgcn_global_load_async_to_lds_b32(
      (g_i32*)(float*)g,
      (l_i32*)l,
      /*offset=*/0, /*cpol=*/0);
#else
  *l = *g;
#endif
}

__device__ __forceinline__ void wait_async0() {
#if defined(__AMDGCN__)
#if __has_builtin(__builtin_amdgcn_s_wait_asynccnt)
  __builtin_amdgcn_s_wait_asynccnt(0);
#else
  asm volatile("s_wait_asynccnt 0" ::: "memory");
#endif
#endif
}

// ---------------------------------------------------------------------------
// Kernel 1: per-batch sequential exclusive prefix over reversed strengths,
// producing weights. Reversed order: v2 (d2), v1 (d1), then S[511]..S[0].
// Loads/stores are coalesced across b (threads = consecutive b).
// ---------------------------------------------------------------------------
__global__ __launch_bounds__(256) void weights_kernel(
    const float* __restrict__ S, const float* __restrict__ u,
    const float* __restrict__ d1, const float* __restrict__ d2,
    float* __restrict__ wvals, float* __restrict__ wv1, float* __restrict__ wv2)
{
  const int b = blockIdx.x * 256 + threadIdx.x;
  if (b >= BATCH) return;

  const float ub = u[b];
  float c = 0.f;

  float s = d2[b];
  wv2[b] = fminf(s, fmaxf(ub - c, 0.f));
  c += s;

  s = d1[b];
  wv1[b] = fminf(s, fmaxf(ub - c, 0.f));
  c += s;

  for (int n = N_SLOTS - 1; n >= 0; --n) {
    s = S[(size_t)n * BATCH + b];
    wvals[(size_t)n * BATCH + b] = fminf(s, fmaxf(ub - c, 0.f));
    c += s;
  }
}

// ---------------------------------------------------------------------------
// Kernel 2: streaming weighted reduction over the slot dimension.
// One block (8 wave32s) per batch b. Weights for this b are staged into LDS
// via CDNA5 async global->LDS DMA (broadcast data: each weight is read by all
// 32 lanes of a wave). The 256 MB `values` stream uses non-temporal b128
// loads so it does not thrash L2 (read-once data).
// ---------------------------------------------------------------------------
__global__ __launch_bounds__(256) void slot_reduce_kernel(
    const float* __restrict__ values, const float* __restrict__ v1,
    const float* __restrict__ v2, const float* __restrict__ wvals,
    const float* __restrict__ wv1, const float* __restrict__ wv2,
    float* __restrict__ out)
{
  __shared__ float w_lds[N_SLOTS];
  __shared__ float psum[8][READ];

  const int b    = blockIdx.x;
  const int t    = threadIdx.x;   // 0..255
  const int wave = t >> 5;        // 0..7
  const int lane = t & 31;        // 0..31

  // Stage this batch's 512 weights into LDS (async DMA on gfx1250).
  async_copy_f32(wvals + (size_t)t * BATCH + b, &w_lds[t]);
  async_copy_f32(wvals + (size_t)(t + 256) * BATCH + b, &w_lds[t + 256]);
#if HAVE_ASYNC_LDS
  wait_async0();
#endif
  __syncthreads();

  // Each wave reduces a 64-slot chunk; each lane owns 4 consecutive r values.
  const int n0 = wave * 64;
  const float* vbase = values + (size_t)b * READ + lane * 4;

  v4f acc = {0.f, 0.f, 0.f, 0.f};
#pragma unroll 8
  for (int n = 0; n < 64; ++n) {
    const v4f* p = (const v4f*)(vbase + (size_t)(n0 + n) * ((size_t)BATCH * READ));
    v4f v = __builtin_nontemporal_load(p);   // global_load_b128 ... nt
    const float w = w_lds[n0 + n];           // ds_load broadcast (uniform)
    acc += w * v;                            // 4x v_fmac_f32
  }

  // Fold in the two appended vectors (order of summation is irrelevant).
  if (wave == 0) {
    v4f v = *(const v4f*)(v1 + (size_t)b * READ + lane * 4);
    acc += wv1[b] * v;
  } else if (wave == 1) {
    v4f v = *(const v4f*)(v2 + (size_t)b * READ + lane * 4);
    acc += wv2[b] * v;
  }

#pragma unroll
  for (int i = 0; i < 4; ++i) psum[wave][lane * 4 + i] = acc[i];
  __syncthreads();

  // Cross-wave reduce (fixed order -> deterministic), full overwrite of out.
  if (t < READ) {
    float ssum = 0.f;
#pragma unroll
    for (int w = 0; w < 8; ++w) ssum += psum[w][t];
    out[(size_t)b * READ + t] = ssum;
  }
}

// ---------------------------------------------------------------------------
extern "C" void kernel_launch(void* const* d_in, const int* in_sizes, int n_in,
                              void* d_out, int out_size, void* d_ws, size_t ws_size,
                              hipStream_t stream) {
  (void)in_sizes; (void)n_in; (void)out_size; (void)ws_size;

  const float* values = (const float*)d_in[0];  // (512,1024,128)
  const float* S      = (const float*)d_in[1];  // (512,1024,1)
  const float* u      = (const float*)d_in[2];  // (1024,1)
  const float* d1     = (const float*)d_in[3];  // (1024,1)
  const float* d2     = (const float*)d_in[4];  // (1024,1)
  const float* v1     = (const float*)d_in[5];  // (1024,128)
  const float* v2     = (const float*)d_in[6];  // (1024,128)
  float* out = (float*)d_out;                   // (1024,128)

  // Workspace layout: 512*1024 slot weights, then wv1[1024], wv2[1024].
  float* wvals = (float*)d_ws;
  float* wv1   = wvals + (size_t)N_SLOTS * BATCH;
  float* wv2   = wv1 + BATCH;

  weights_kernel<<<dim3((BATCH + 255) / 256), dim3(256), 0, stream>>>(
      S, u, d1, d2, wvals, wv1, wv2);

  slot_reduce_kernel<<<dim3(BATCH), dim3(256), 0, stream>>>(
      values, v1, v2, wvals, wv1, wv2, out);
}